// MultiModalFusion_67516885893470
// MI455X (gfx1250) — compile-verified
//
#include <hip/hip_runtime.h>

// Problem constants from the reference
#define B_   4
#define N_   51
#define M_   4
#define S_   256
#define DK_  64
#define DM_  128

typedef float v2f __attribute__((ext_vector_type(2)));
typedef float v8f __attribute__((ext_vector_type(8)));

// ---------------------------------------------------------------------------
// Pre-kernel: fold the projection weights into 4 scalars.
//   A  = dot(Wq.sum(1), Wk.sum(1))     (coefficient of xq*xk in scores)
//   Cc = dot(bq, Wk.sum(1))            (coefficient of xk; survives softmax)
//   mv = mean_k(Wv.sum(1))             (output scale)
//   mb = mean_k(bv)                    (output bias)
// (The xq-only and constant score terms cancel in softmax over t.)
// ---------------------------------------------------------------------------
__global__ __launch_bounds__(64)
void mmf_scalars(const float* __restrict__ Wq, const float* __restrict__ bq,
                 const float* __restrict__ Wk, const float* __restrict__ Wv,
                 const float* __restrict__ bv, float* __restrict__ ws) {
  __shared__ float wqs[DK_], wks[DK_], wvs[DK_];
  const int k = threadIdx.x;          // 0..63
  float sq = 0.f, sk = 0.f, sv = 0.f;
  for (int d = 0; d < DM_; ++d) {
    sq += Wq[k * DM_ + d];
    sk += Wk[k * DM_ + d];
    sv += Wv[k * DM_ + d];
  }
  wqs[k] = sq; wks[k] = sk; wvs[k] = sv;
  __syncthreads();
  if (k == 0) {
    float A = 0.f, Cc = 0.f, mv = 0.f, mb = 0.f;
    for (int t = 0; t < DK_; ++t) {
      A  += wqs[t] * wks[t];
      Cc += bq[t]  * wks[t];
      mv += wvs[t];
      mb += bv[t];
    }
    ws[0] = A;
    ws[1] = Cc;
    ws[2] = mv * (1.0f / DK_);
    ws[3] = mb * (1.0f / DK_);
  }
}

// ---------------------------------------------------------------------------
// Main kernel: one block per (b,n). 8 waves, 64 tasks = (modal i) x (s-tile).
// Per task & target modal j: E[s,t] = exp(alpha_s * x_j[t] - rowmax_s),
// reduced over t via V_WMMA_F32_16X16X4_F32 against B = [ones | x_j | 0...].
// C column 0 accumulates rowsum(E); column 1 accumulates rowsum(E*x_j).
// out[b,n,i,s] = mb + mv * mean_j( wsum/rsum ).
// ---------------------------------------------------------------------------
__global__ __launch_bounds__(256)
void mmf_main(const float* __restrict__ x, const float* __restrict__ ws,
              float* __restrict__ out) {
  __shared__ float xs[M_][S_];
  __shared__ float xmx[M_], xmn[M_];
  __shared__ float stage[8][32];      // per-wave: [0..15]=rowsum, [16..31]=weighted

  const int bid  = blockIdx.x;        // b*N + n
  const int tid  = threadIdx.x;
  const int wv   = tid >> 5;          // wave id (wave32)
  const int lane = tid & 31;

  // Stage x[b,n,:,:] (4x256 f32 = 4KB) into LDS.
  const float* xb = x + (size_t)bid * (M_ * S_);
#pragma unroll
  for (int mm = 0; mm < M_; ++mm) xs[mm][tid] = xb[mm * S_ + tid];
  __syncthreads();

  // Per-modal min/max for the softmax shift (one wave per modal).
  if (wv < M_) {
    float mx = -1e30f, mn = 1e30f;
    for (int t = lane; t < S_; t += 32) {
      const float v = xs[wv][t];
      mx = fmaxf(mx, v);
      mn = fminf(mn, v);
    }
#pragma unroll
    for (int off = 16; off > 0; off >>= 1) {
      mx = fmaxf(mx, __shfl_xor(mx, off, 32));
      mn = fminf(mn, __shfl_xor(mn, off, 32));
    }
    if (lane == 0) { xmx[wv] = mx; xmn[wv] = mn; }
  }
  __syncthreads();

  const float Ascal = ws[0];
  const float Cc    = ws[1];
  const float mv    = ws[2];
  const float mb    = ws[3];
  const float scale = 0.125f;         // 1/sqrt(64)

  const int half = lane >> 4;         // A-matrix: lanes 0-15 -> K 0,1 ; 16-31 -> K 2,3
  const int m    = lane & 15;         // row within the 16-row s-tile / B column id

  for (int task = wv; task < 64; task += 8) {
    const int   i     = task >> 4;
    const int   s0    = (task & 15) << 4;
    const float xi    = xs[i][s0 + m];
    const float alpha = (Ascal * xi + Cc) * scale;
    float acc = 0.f;

#pragma unroll
    for (int jj = 0; jj < M_ - 1; ++jj) {
      const int   j    = jj + (jj >= i);
      const float rmax = (alpha >= 0.f) ? alpha * xmx[j] : alpha * xmn[j];

      v8f c = {0.f, 0.f, 0.f, 0.f, 0.f, 0.f, 0.f, 0.f};

#pragma unroll 8
      for (int t0 = 0; t0 < S_; t0 += 4) {
        // A-matrix tile: E[s0+m, t0 + 2*half + {0,1}]
        const float xa0 = xs[j][t0 + 2 * half + 0];
        const float xa1 = xs[j][t0 + 2 * half + 1];
        v2f a;
        a.x = __expf(alpha * xa0 - rmax);
        a.y = __expf(alpha * xa1 - rmax);
        // B-matrix: rows K = 2*half + {0,1}, column n = m.
        //   col 0 = ones (rowsum), col 1 = x_j[t] (weighted rowsum), rest 0.
        v2f bm;
        bm.x = (m == 0) ? 1.0f : ((m == 1) ? xa0 : 0.0f);
        bm.y = (m == 0) ? 1.0f : ((m == 1) ? xa1 : 0.0f);
        c = __builtin_amdgcn_wmma_f32_16x16x4_f32(
                false, a, false, bm, (short)0, c, false, false);
      }

      // C/D layout: col n = lane%16; rows v + 8*(lane/16).
      // Lanes 0/16 hold the 16 rowsums, lanes 1/17 the weighted sums.
      if (m <= 1) {
#pragma unroll
        for (int v = 0; v < 8; ++v)
          stage[wv][16 * m + 8 * half + v] = c[v];
      }
      __syncthreads();   // uniform trip counts across all waves -> convergent
      const float rs   = stage[wv][m];
      const float wsum = stage[wv][16 + m];
      acc += wsum / rs;  // rs >= 1 (max-shifted exp)
      __syncthreads();
    }

    if (lane < 16)
      out[((size_t)bid * M_ + i) * S_ + s0 + m] = mb + mv * (1.0f / 3.0f) * acc;
  }
}

// ---------------------------------------------------------------------------
extern "C" void kernel_launch(void* const* d_in, const int* in_sizes, int n_in,
                              void* d_out, int out_size, void* d_ws, size_t ws_size,
                              hipStream_t stream) {
  const float* x  = (const float*)d_in[0];
  const float* Wq = (const float*)d_in[1];
  const float* bq = (const float*)d_in[2];
  const float* Wk = (const float*)d_in[3];
  // d_in[4] = bk: unused (its score terms are constant in t and cancel in softmax)
  const float* Wv = (const float*)d_in[5];
  const float* bv = (const float*)d_in[6];
  float* ws = (float*)d_ws;   // 4 scalars

  mmf_scalars<<<1, 64, 0, stream>>>(Wq, bq, Wk, Wv, bv, ws);
  mmf_main<<<B_ * N_, 256, 0, stream>>>(x, ws, (float*)d_out);
}